// SimpleHGN_MTL_Layer_18580028522791
// MI455X (gfx1250) — compile-verified
//
#include <hip/hip_runtime.h>
#include <hip/hip_bf16.h>

#define N_HEAD   100000
#define N_TAIL   100000
#define E_EDGES  1600000
#define IN_F     128
#define OUT_F    32
#define NHEADS   4
#define HID      128     // NHEADS * OUT_F
#define N_ETYPES 8
#define EDGE_F   32
#define SLOPE    0.2f

typedef float v2f __attribute__((ext_vector_type(2)));
typedef float v8f __attribute__((ext_vector_type(8)));

// ---------- order-preserving float <-> uint encoding for atomic max ----------
__device__ __forceinline__ unsigned enc_f32(float f) {
    int i = __float_as_int(f);
    return (i >= 0) ? ((unsigned)i | 0x80000000u) : ~(unsigned)i;
}
__device__ __forceinline__ float dec_f32(unsigned u) {
    int i = (u & 0x80000000u) ? (int)(u & 0x7FFFFFFFu) : (int)(~u);
    return __int_as_float(i);
}

// =====================================================================
// Kernel A: fused H = X @ W for head and tail features, fp32 WMMA.
// One wave -> one 16-row tile, full 128-wide output.
// W kept in LDS, permuted so each 4x16 B fragment is a single ds_load_b64:
//   Wl[(k>>1)*256 + n*2 + (k&1)] = W[k*128 + n]
// =====================================================================
__global__ __launch_bounds__(256) void gemm_wmma_f32(
    const float* __restrict__ head, const float* __restrict__ tail,
    const float* __restrict__ W,
    float* __restrict__ Hhead, float* __restrict__ Htail)
{
    __shared__ float Wl[IN_F * HID];   // 64 KB

    for (int idx = threadIdx.x; idx < IN_F * HID; idx += 256) {
        int k = idx >> 7, n = idx & 127;
        Wl[(k >> 1) * 256 + n * 2 + (k & 1)] = W[idx];
    }
    __syncthreads();

    const int wave = threadIdx.x >> 5;
    const int lane = threadIdx.x & 31;
    const int tile = blockIdx.x * 8 + wave;
    const int n_tiles = (N_HEAD + N_TAIL) / 16;        // 12500
    if (tile >= n_tiles) return;                       // wave-uniform

    const float* src;
    float*       dst;
    int row0;
    if (tile < N_HEAD / 16) { src = head; dst = Hhead; row0 = tile * 16; }
    else                    { src = tail; dst = Htail; row0 = (tile - N_HEAD / 16) * 16; }

    const int m  = lane & 15;
    const int hi = lane >> 4;                          // 0 or 1
    const float* arow = src + (size_t)(row0 + m) * IN_F + 2 * hi;

    v8f acc[8] = {};

    for (int k0 = 0; k0 < IN_F; k0 += 4) {
        // A fragment (16x4): lane holds A[m][k0+2*hi], A[m][k0+2*hi+1]
        v2f a = *(const v2f*)(arow + k0);
        const int krow = (k0 >> 1) + hi;               // K-pair row in Wl
        const float* bbase = &Wl[krow * 256];
#pragma unroll
        for (int t = 0; t < 8; t++) {
            // B fragment (4x16): lane holds B[k0+2*hi][t*16+m], B[k0+2*hi+1][t*16+m]
            v2f b = *(const v2f*)(bbase + (t * 16 + m) * 2);
            acc[t] = __builtin_amdgcn_wmma_f32_16x16x4_f32(
                false, a, false, b, (short)0, acc[t], false, false);
        }
    }

    // D layout: VGPR r, lanes 0-15 -> M=r, lanes 16-31 -> M=8+r; N = t*16 + m
#pragma unroll
    for (int t = 0; t < 8; t++) {
        const int n = t * 16 + m;
#pragma unroll
        for (int r = 0; r < 8; r++) {
            const int row = row0 + r + 8 * hi;
            dst[(size_t)row * HID + n] = acc[t][r];
        }
    }
}

// =====================================================================
// Kernel B1: h_l[node,h] = sum_d a_l[h,d] * Hhead[node, h*32+d]   (and h_r)
// =====================================================================
__global__ __launch_bounds__(256) void node_scores(
    const float* __restrict__ Hhead, const float* __restrict__ Htail,
    const float* __restrict__ a_l,   const float* __restrict__ a_r,
    float* __restrict__ h_l, float* __restrict__ h_r)
{
    int pair = blockIdx.x * 256 + threadIdx.x;
    const int per_side = N_HEAD * NHEADS;              // 400000
    if (pair >= 2 * per_side) return;
    const bool is_tail = pair >= per_side;
    const int p = is_tail ? pair - per_side : pair;
    const int node = p >> 2, h = p & 3;
    const float* H = is_tail ? Htail : Hhead;
    const float* a = (is_tail ? a_r : a_l) + h * OUT_F;
    const float* row = H + (size_t)node * HID + h * OUT_F;
    float s = 0.f;
#pragma unroll 8
    for (int d = 0; d < OUT_F; d++) s = fmaf(a[d], row[d], s);
    (is_tail ? h_r : h_l)[p] = s;
}

// =====================================================================
// Kernel B2: per-edge-type scores: e = edge_emb @ W_e; h_e[t,h] = sum(a_e * e)
// =====================================================================
__global__ __launch_bounds__(256) void etype_scores(
    const float* __restrict__ edge_emb, const float* __restrict__ W_e,
    const float* __restrict__ a_e, float* __restrict__ h_e)
{
    __shared__ float ebuf[N_ETYPES * (EDGE_F * NHEADS)];   // 8 x 128
    for (int idx = threadIdx.x; idx < N_ETYPES * 128; idx += 256) {
        int t = idx >> 7, c = idx & 127;
        float s = 0.f;
        for (int k = 0; k < EDGE_F; k++)
            s = fmaf(edge_emb[t * EDGE_F + k], W_e[k * 128 + c], s);
        ebuf[idx] = s;
    }
    __syncthreads();
    if (threadIdx.x < N_ETYPES * NHEADS) {
        int t = threadIdx.x >> 2, h = threadIdx.x & 3;
        float s = 0.f;
        for (int d = 0; d < EDGE_F; d++)
            s = fmaf(a_e[h * EDGE_F + d], ebuf[t * 128 + h * EDGE_F + d], s);
        h_e[threadIdx.x] = s;
    }
}

// =====================================================================
// Kernel C1: att = leakyrelu(h_l[head] + h_r[tail] + h_e[etype]); segment max
// =====================================================================
__global__ __launch_bounds__(256) void attn_logits(
    const int* __restrict__ heads, const int* __restrict__ tails,
    const int* __restrict__ etype,
    const float* __restrict__ h_l, const float* __restrict__ h_r,
    const float* __restrict__ h_e,
    float* __restrict__ att, unsigned* __restrict__ m_enc)
{
    int e = blockIdx.x * 256 + threadIdx.x;
    if (e >= E_EDGES) return;
    const int hd = heads[e], tl = tails[e], et = etype[e];
    float4 l  = *(const float4*)(h_l + hd * 4);
    float4 r  = *(const float4*)(h_r + tl * 4);
    float4 ee = *(const float4*)(h_e + et * 4);
    float v[4] = { l.x + r.x + ee.x, l.y + r.y + ee.y,
                   l.z + r.z + ee.z, l.w + r.w + ee.w };
#pragma unroll
    for (int h = 0; h < 4; h++) {
        float a = v[h];
        a = a > 0.f ? a : SLOPE * a;
        att[(size_t)e * 4 + h] = a;
        atomicMax(&m_enc[hd * 4 + h], enc_f32(a));
    }
}

// =====================================================================
// Kernel C2: att_exp = exp(att - m[head]); segment sum of denominators
// =====================================================================
__global__ __launch_bounds__(256) void attn_exp(
    const int* __restrict__ heads,
    const unsigned* __restrict__ m_enc,
    float* __restrict__ att, float* __restrict__ denom)
{
    int e = blockIdx.x * 256 + threadIdx.x;
    if (e >= E_EDGES) return;
    const int hd = heads[e];
#pragma unroll
    for (int h = 0; h < 4; h++) {
        float m  = dec_f32(m_enc[hd * 4 + h]);
        float ex = __expf(att[(size_t)e * 4 + h] - m);
        att[(size_t)e * 4 + h] = ex;
        atomicAdd(&denom[hd * 4 + h], ex);
    }
}

// =====================================================================
// Kernel C3: out[head] += (att_exp/denom) * Htail[tail]   (wave per edge)
// lane covers float4 at column lane*4; head index = lane/8
// =====================================================================
__global__ __launch_bounds__(256) void aggregate(
    const int* __restrict__ heads, const int* __restrict__ tails,
    const float* __restrict__ att_exp, const float* __restrict__ denom,
    const float* __restrict__ Htail, float* __restrict__ out)
{
    const int e = blockIdx.x * 8 + (threadIdx.x >> 5);
    if (e >= E_EDGES) return;
    const int lane = threadIdx.x & 31;
    const int hd = heads[e], tl = tails[e];
    const int h = lane >> 3;
    const float w = att_exp[(size_t)e * 4 + h] / denom[hd * 4 + h];
    float4 x = *(const float4*)(Htail + (size_t)tl * HID + lane * 4);
    float* o = out + (size_t)hd * HID + lane * 4;
    atomicAdd(o + 0, w * x.x);
    atomicAdd(o + 1, w * x.y);
    atomicAdd(o + 2, w * x.z);
    atomicAdd(o + 3, w * x.w);
}

// =====================================================================
// Kernel D: in-place ELU
// =====================================================================
__global__ __launch_bounds__(256) void elu_inplace(float* __restrict__ out, int n)
{
    int i = blockIdx.x * 256 + threadIdx.x;
    if (i >= n) return;
    float x = out[i];
    out[i] = x > 0.f ? x : (__expf(x) - 1.f);
}

// =====================================================================
extern "C" void kernel_launch(void* const* d_in, const int* in_sizes, int n_in,
                              void* d_out, int out_size, void* d_ws, size_t ws_size,
                              hipStream_t stream) {
    const float* head_feature = (const float*)d_in[0];
    const float* tail_feature = (const float*)d_in[1];
    const int*   edge_list    = (const int*)d_in[2];   // [2, E]
    const int*   tmp_edge     = (const int*)d_in[3];   // [E]
    const float* W            = (const float*)d_in[4];
    const float* W_e          = (const float*)d_in[5];
    const float* a_l          = (const float*)d_in[6];
    const float* a_r          = (const float*)d_in[7];
    const float* a_e          = (const float*)d_in[8];
    const float* edge_emb     = (const float*)d_in[9];
    float* out = (float*)d_out;

    const int* heads = edge_list;
    const int* tails = edge_list + E_EDGES;

    char* ws = (char*)d_ws;
    size_t off = 0;
    auto alloc = [&](size_t bytes) -> void* {
        void* p = ws + off;
        off += (bytes + 255) & ~(size_t)255;
        return p;
    };
    float*    Hhead  = (float*)alloc((size_t)N_HEAD * HID * 4);
    float*    Htail  = (float*)alloc((size_t)N_TAIL * HID * 4);
    float*    h_l    = (float*)alloc((size_t)N_HEAD * NHEADS * 4);
    float*    h_r    = (float*)alloc((size_t)N_TAIL * NHEADS * 4);
    float*    h_e    = (float*)alloc((size_t)N_ETYPES * NHEADS * 4);
    unsigned* m_enc  = (unsigned*)alloc((size_t)N_HEAD * NHEADS * 4);
    float*    denom  = (float*)alloc((size_t)N_HEAD * NHEADS * 4);
    float*    att    = (float*)alloc((size_t)E_EDGES * NHEADS * 4);

    hipMemsetAsync(m_enc, 0, (size_t)N_HEAD * NHEADS * 4, stream);
    hipMemsetAsync(denom, 0, (size_t)N_HEAD * NHEADS * 4, stream);
    hipMemsetAsync(out,   0, (size_t)N_HEAD * HID * 4, stream);

    // A: fused GEMMs (WMMA fp32)
    {
        const int n_tiles = (N_HEAD + N_TAIL) / 16;          // 12500
        const int blocks  = (n_tiles + 7) / 8;
        gemm_wmma_f32<<<blocks, 256, 0, stream>>>(head_feature, tail_feature, W,
                                                  Hhead, Htail);
    }
    // B1: per-node attention scores
    {
        const int total = 2 * N_HEAD * NHEADS;
        node_scores<<<(total + 255) / 256, 256, 0, stream>>>(Hhead, Htail, a_l, a_r,
                                                             h_l, h_r);
    }
    // B2: per-edge-type scores
    etype_scores<<<1, 256, 0, stream>>>(edge_emb, W_e, a_e, h_e);

    // C1: logits + segment max
    attn_logits<<<(E_EDGES + 255) / 256, 256, 0, stream>>>(heads, tails, tmp_edge,
                                                           h_l, h_r, h_e, att, m_enc);
    // C2: exp + segment sum
    attn_exp<<<(E_EDGES + 255) / 256, 256, 0, stream>>>(heads, m_enc, att, denom);

    // C3: weighted scatter-aggregate (wave per edge)
    aggregate<<<(E_EDGES + 7) / 8, 256, 0, stream>>>(heads, tails, att, denom,
                                                     Htail, out);
    // D: ELU
    {
        const int n = N_HEAD * HID;
        elu_inplace<<<(n + 255) / 256, 256, 0, stream>>>(out, n);
    }
}